// FlexibleDisentangledMultiheadAttention_50672024158410
// MI455X (gfx1250) — compile-verified
//
#include <hip/hip_runtime.h>
#include <math.h>

// Problem constants (match reference)
#define BD    4
#define SD    512
#define IND   768
#define EMBD  768
#define NH    12
#define HD    64
#define RLEN  1023   // 2*MAX_LEN - 1

typedef __attribute__((ext_vector_type(2))) float v2f;
typedef __attribute__((ext_vector_type(8))) float v8f;

// fp32 WMMA: D(16x16) = A(16x4) * B(4x16) + C, wave32.
__device__ __forceinline__ v8f wmma_f32(v2f a, v2f b, v8f c) {
  return __builtin_amdgcn_wmma_f32_16x16x4_f32(
      /*neg_a=*/false, a, /*neg_b=*/false, b,
      /*c_mod=*/(short)0, c, /*reuse_a=*/false, /*reuse_b=*/false);
}

// Generic C[M,N] = A[M,K] @ Bw[K,N] + bias[N].
// One wave per block; wave computes 16 rows x 64 cols (4 C tiles).
// mode 0: plain row-major store. mode 1: split-head store [B,H,S,D].
__global__ void deberta_gemm_f32(const float* __restrict__ A,
                                 const float* __restrict__ Bw,
                                 const float* __restrict__ bias,
                                 float* __restrict__ Cout,
                                 int M, int N, int K, int mode) {
  const int lane = threadIdx.x & 31;
  const int half = lane >> 4;     // selects K pair within 4-wide chunk
  const int mr   = lane & 15;     // row (A) / col (B) within tile
  const int n0   = blockIdx.x * 64;
  const int m0   = blockIdx.y * 16;

  v8f acc[4] = {};

  const int  row   = m0 + mr;
  const bool rowOK = row < M;
  const float* arow = A + (size_t)row * K;

  for (int kk = 0; kk < K; kk += 4) {
    v2f a;
    if (rowOK) a = *(const v2f*)(arow + kk + 2 * half);
    else       { a.x = 0.f; a.y = 0.f; }
#pragma unroll
    for (int c = 0; c < 4; ++c) {
      const int n = n0 + 16 * c + mr;
      v2f b;
      b.x = Bw[(size_t)(kk + 2 * half)     * N + n];
      b.y = Bw[(size_t)(kk + 2 * half + 1) * N + n];
      acc[c] = wmma_f32(a, b, acc[c]);
    }
  }

#pragma unroll
  for (int c = 0; c < 4; ++c) {
#pragma unroll
    for (int rr = 0; rr < 8; ++rr) {
      const int m = m0 + rr + 8 * half;   // C layout: lanes16-31 hold M=rr+8
      const int n = n0 + 16 * c + mr;
      if (m < M) {
        const float val = acc[c][rr] + (bias ? bias[n] : 0.f);
        if (mode == 0) {
          Cout[(size_t)m * N + n] = val;
        } else {                           // split heads -> [B,H,S,D]
          const int b_ = m / SD, i = m % SD, h = n / HD, d = n % HD;
          Cout[(((size_t)b_ * NH + h) * SD + i) * HD + d] = val;
        }
      }
    }
  }
}

// scores[bh,i,j] = q_i·k_j + q_i·pk[i-j+511] + pq[i-j+511]·k_j, all via fp32 WMMA.
// Relative terms done as shifted GEMMs over t = (i-j+15)-(i0-j0):
//   Ctmp[m,t] = q[i0+m]·pk[r0+t]   (shares A with q·k^T)
//   Dtmp[t,n] = pq[r0+t]·k[j0+n]   (shares B with q·k^T)
// with r0 = i0-j0+496; then score(m,n) += Ctmp[m,m-n+15] + Dtmp[m-n+15,n]
// gathered through LDS. One wave per 16x16 score tile, 80 WMMAs/tile.
__global__ void deberta_scores(const float* __restrict__ Q,
                               const float* __restrict__ Km,
                               const float* __restrict__ PK,
                               const float* __restrict__ PQ,
                               float* __restrict__ SC) {
  __shared__ float cbuf[16 * 32];   // Ctmp[m][t]
  __shared__ float dbuf[32 * 16];   // Dtmp[t][n]

  const int lane = threadIdx.x & 31;
  const int half = lane >> 4;
  const int mr   = lane & 15;
  const int j0   = blockIdx.x * 16;
  const int i0   = blockIdx.y * 16;
  const int bh   = blockIdx.z;
  const int h    = bh % NH;
  const int r0   = i0 - j0 + 496;   // >= 0; r0+30 <= 1022 (t=31 unused)

  const float* q = Q  + (size_t)bh * SD * HD;
  const float* k = Km + (size_t)bh * SD * HD;

  const float* qrow = q + (size_t)(i0 + mr) * HD;
  const float* krow = k + (size_t)(j0 + mr) * HD;
  const int rA = min(r0 + mr,      RLEN - 1);  // pk/pq row, t-tile 0 (clamp hits only t=31)
  const int rB = min(r0 + 16 + mr, RLEN - 1);  // t-tile 1
  const float* pkA = PK + (size_t)rA * EMBD + h * HD;
  const float* pkB = PK + (size_t)rB * EMBD + h * HD;
  const float* pqA = PQ + (size_t)rA * EMBD + h * HD;
  const float* pqB = PQ + (size_t)rB * EMBD + h * HD;

  v8f qk = {}, cpA = {}, cpB = {}, pcA = {}, pcB = {};
#pragma unroll
  for (int kk = 0; kk < HD; kk += 4) {
    const int o = kk + 2 * half;
    const v2f aq  = *(const v2f*)(qrow + o);
    const v2f bk_ = *(const v2f*)(krow + o);
    const v2f bkA = *(const v2f*)(pkA + o);
    const v2f bkB = *(const v2f*)(pkB + o);
    const v2f aqA = *(const v2f*)(pqA + o);
    const v2f aqB = *(const v2f*)(pqB + o);
    qk  = wmma_f32(aq,  bk_, qk);    // q · k^T
    cpA = wmma_f32(aq,  bkA, cpA);   // c2p, t = 0..15
    cpB = wmma_f32(aq,  bkB, cpB);   // c2p, t = 16..31
    pcA = wmma_f32(aqA, bk_, pcA);   // p2c, t = 0..15
    pcB = wmma_f32(aqB, bk_, pcB);   // p2c, t = 16..31
  }

  // Spill relative tiles to LDS for the diagonal gather.
  // C layout: VGPR rr, lanes 0-15 -> M=rr, lanes 16-31 -> M=rr+8; N=lane&15.
#pragma unroll
  for (int rr = 0; rr < 8; ++rr) {
    const int m = rr + 8 * half;               // Ctmp row / Dtmp t (within tile)
    cbuf[m * 32 + mr]        = cpA[rr];
    cbuf[m * 32 + 16 + mr]   = cpB[rr];
    dbuf[m * 16 + mr]        = pcA[rr];
    dbuf[(m + 16) * 16 + mr] = pcB[rr];
  }
  __syncthreads();

  float* out = SC + (size_t)bh * SD * SD;
  const int n = mr;
#pragma unroll
  for (int rr = 0; rr < 8; ++rr) {
    const int m = rr + 8 * half;
    const int t = m - n + 15;                  // in [0,30]
    out[(size_t)(i0 + m) * SD + (j0 + n)] =
        qk[rr] + cbuf[m * 32 + t] + dbuf[t * 16 + n];
  }
}

// In-place softmax over the last dim (512), with 1/sqrt(3D) scaling.
// One 256-thread block per row, 2 elements per thread.
__global__ void deberta_softmax(float* __restrict__ SC) {
  __shared__ float red[256];
  const int t = threadIdx.x;
  float* row = SC + (size_t)blockIdx.x * SD;
  const float scale = 0.07216878364870322f;   // 1/sqrt(64*3)

  float s0 = row[t] * scale;
  float s1 = row[t + 256] * scale;

  red[t] = fmaxf(s0, s1);
  __syncthreads();
  for (int off = 128; off > 0; off >>= 1) {
    if (t < off) red[t] = fmaxf(red[t], red[t + off]);
    __syncthreads();
  }
  const float m = red[0];
  __syncthreads();

  const float e0 = __expf(s0 - m), e1 = __expf(s1 - m);
  red[t] = e0 + e1;
  __syncthreads();
  for (int off = 128; off > 0; off >>= 1) {
    if (t < off) red[t] += red[t + off];
    __syncthreads();
  }
  const float inv = 1.f / red[0];
  row[t]       = e0 * inv;
  row[t + 256] = e1 * inv;
}

// ctx[b,i,h*64+d] = sum_j attn[bh,i,j] * v[bh,j,d]
// One wave per 16-row strip; D=64 covered by the wave's 4 C tiles.
__global__ void deberta_av(const float* __restrict__ ATT,
                           const float* __restrict__ V,
                           float* __restrict__ CTX) {
  const int lane = threadIdx.x & 31;
  const int half = lane >> 4;
  const int mr   = lane & 15;
  const int i0   = blockIdx.x * 16;
  const int bh   = blockIdx.y;
  const int b    = bh / NH, h = bh % NH;

  const float* att = ATT + (size_t)bh * SD * SD;
  const float* v   = V   + (size_t)bh * SD * HD;

  v8f acc[4] = {};
  const float* arow = att + (size_t)(i0 + mr) * SD;
  for (int kk = 0; kk < SD; kk += 4) {
    v2f a = *(const v2f*)(arow + kk + 2 * half);
#pragma unroll
    for (int c = 0; c < 4; ++c) {
      const int n = 16 * c + mr;
      v2f bv;
      bv.x = v[(size_t)(kk + 2 * half)     * HD + n];
      bv.y = v[(size_t)(kk + 2 * half + 1) * HD + n];
      acc[c] = wmma_f32(a, bv, acc[c]);
    }
  }

#pragma unroll
  for (int c = 0; c < 4; ++c) {
#pragma unroll
    for (int rr = 0; rr < 8; ++rr) {
      const int i = i0 + rr + 8 * half;
      const int d = 16 * c + mr;
      CTX[((size_t)b * SD + i) * EMBD + h * HD + d] = acc[c][rr];
    }
  }
}

extern "C" void kernel_launch(void* const* d_in, const int* in_sizes, int n_in,
                              void* d_out, int out_size, void* d_ws, size_t ws_size,
                              hipStream_t stream) {
  const float* x    = (const float*)d_in[0];
  const float* Wq   = (const float*)d_in[1];
  const float* bq   = (const float*)d_in[2];
  const float* Wk   = (const float*)d_in[3];
  const float* bk   = (const float*)d_in[4];
  const float* Wv   = (const float*)d_in[5];
  const float* bv   = (const float*)d_in[6];
  const float* rel  = (const float*)d_in[7];
  const float* Wpk  = (const float*)d_in[8];
  const float* bpk  = (const float*)d_in[9];
  const float* Wpq  = (const float*)d_in[10];
  const float* bpq  = (const float*)d_in[11];
  const float* Wo   = (const float*)d_in[12];
  const float* bo   = (const float*)d_in[13];
  float* out = (float*)d_out;

  // Workspace layout (floats): Q,K,V [B,H,S,D]; PK,PQ [RLEN,EMB]; SC [BH,S,S]; CTX [B,S,EMB]
  float* ws  = (float*)d_ws;
  const size_t qkvN = (size_t)BD * NH * SD * HD;      // 1.57M
  const size_t posN = (size_t)RLEN * EMBD;            // 0.79M
  const size_t scN  = (size_t)BD * NH * SD * SD;      // 12.6M
  float* Q   = ws;
  float* K   = Q  + qkvN;
  float* V   = K  + qkvN;
  float* PK  = V  + qkvN;
  float* PQ  = PK + posN;
  float* SC  = PQ + posN;
  float* CTX = SC + scN;

  const dim3 wave(32);
  const int M = BD * SD;  // 2048

  // Projections (fp32 WMMA GEMMs). mode 1 = split-head store.
  deberta_gemm_f32<<<dim3(EMBD / 64, M / 16), wave, 0, stream>>>(x, Wq, bq, Q, M, EMBD, IND, 1);
  deberta_gemm_f32<<<dim3(EMBD / 64, M / 16), wave, 0, stream>>>(x, Wk, bk, K, M, EMBD, IND, 1);
  deberta_gemm_f32<<<dim3(EMBD / 64, M / 16), wave, 0, stream>>>(x, Wv, bv, V, M, EMBD, IND, 1);
  deberta_gemm_f32<<<dim3(EMBD / 64, (RLEN + 15) / 16), wave, 0, stream>>>(rel, Wpk, bpk, PK, RLEN, EMBD, IND, 0);
  deberta_gemm_f32<<<dim3(EMBD / 64, (RLEN + 15) / 16), wave, 0, stream>>>(rel, Wpq, bpq, PQ, RLEN, EMBD, IND, 0);

  // Attention scores: all three terms via WMMA (shifted-GEMM for relative terms)
  deberta_scores<<<dim3(SD / 16, SD / 16, BD * NH), wave, 0, stream>>>(Q, K, PK, PQ, SC);

  // Softmax (in place)
  deberta_softmax<<<dim3(BD * NH * SD), dim3(256), 0, stream>>>(SC);

  // attn @ v -> ctx [B,S,EMB]
  deberta_av<<<dim3(SD / 16, BD * NH), wave, 0, stream>>>(SC, V, CTX);

  // Output projection straight into d_out
  deberta_gemm_f32<<<dim3(IND / 64, M / 16), wave, 0, stream>>>(CTX, Wo, bo, out, M, IND, EMBD, 0);
}